// GRU_21045339750966
// MI455X (gfx1250) — compile-verified
//
#include <hip/hip_runtime.h>

// 2-layer GRU (H=10) fused recurrence for gfx1250 using V_WMMA_F32_16X16X4_F32.
// One wave = 16 batch rows. LDS row per batch m (24 floats):
//   [ h1(0..9), x_t(10), 1.0(11), h2(12..21), pad(22..23) ]
// WMMA A-operand chunks are read straight from these rows; x and bias terms are
// folded into the matmul via the extra K slots. Gate tiles keep each gate type
// at lanes 0..9 so r/z/n/h_prev stay lane-aligned (no cross-lane ops).
// v2: x loads software-pipelined 2 steps ahead; layer-2 h2-side WMMAs hoisted
// ahead of layer-1 gate math (they only need the previous step's h2); duplicate
// same-address stores replace exec-mask divergence for x staging / output.

typedef float v2f __attribute__((ext_vector_type(2)));
typedef float v8f __attribute__((ext_vector_type(8)));

#define BB   1024
#define TT   2048
#define HN   10
#define ROWW 24
#define L2E  1.4426950408889634f

static __device__ __forceinline__ v8f wmma4(v2f a, v2f b, v8f c) {
  // D = A(16x4) * B(4x16) + C, fp32
  return __builtin_amdgcn_wmma_f32_16x16x4_f32(false, a, false, b, (short)0, c, false, false);
}

// B-operand element for layer 1.  K: 0..9 = h1, 10 = x_t, 11 = 1.0
// tile 0: r sums (scaled log2e)  tile 1: z sums (log2e)
// tile 2: hn (scaled 2*log2e)    tile 3: xn (2*log2e)
static __device__ float brow1(int k, int c, int tile,
                              const float* wih, const float* whh,
                              const float* bih, const float* bhh) {
  if (c >= HN) return 0.0f;
  if (tile <= 1) {
    int g = (tile == 0) ? c : HN + c;
    if (k < HN)  return whh[g * HN + k] * L2E;
    if (k == 10) return wih[g] * L2E;                  // w_ih0 is [3H,1]
    if (k == 11) return (bih[g] + bhh[g]) * L2E;
  } else if (tile == 2) {
    int g = 2 * HN + c;
    if (k < HN)  return whh[g * HN + k] * (2.0f * L2E);
    if (k == 11) return bhh[g] * (2.0f * L2E);
  } else {
    int g = 2 * HN + c;
    if (k == 10) return wih[g] * (2.0f * L2E);
    if (k == 11) return bih[g] * (2.0f * L2E);
  }
  return 0.0f;
}

// Layer 2.  K: 0..9 = h1_new, 10 = unused, 11 = 1.0, 12..21 = h2, 22..23 = 0
static __device__ float brow2(int k, int c, int tile,
                              const float* wih, const float* whh,
                              const float* bih, const float* bhh) {
  if (c >= HN) return 0.0f;
  if (tile <= 1) {
    int g = (tile == 0) ? c : HN + c;
    if (k < HN)            return wih[g * HN + k] * L2E;
    if (k == 11)           return (bih[g] + bhh[g]) * L2E;
    if (k >= 12 && k < 22) return whh[g * HN + (k - 12)] * L2E;
  } else if (tile == 2) {
    int g = 2 * HN + c;
    if (k >= 12 && k < 22) return whh[g * HN + (k - 12)] * (2.0f * L2E);
    if (k == 11)           return bhh[g] * (2.0f * L2E);
  } else {
    int g = 2 * HN + c;
    if (k < HN)            return wih[g * HN + k] * (2.0f * L2E);
    if (k == 11)           return bih[g] * (2.0f * L2E);
  }
  return 0.0f;
}

__global__ __launch_bounds__(32) void gru2_wmma(
    const float* __restrict__ x,
    const float* __restrict__ w_ih0, const float* __restrict__ w_hh0,
    const float* __restrict__ b_ih0, const float* __restrict__ b_hh0,
    const float* __restrict__ w_ih1, const float* __restrict__ w_hh1,
    const float* __restrict__ b_ih1, const float* __restrict__ b_hh1,
    const float* __restrict__ w_lin, const float* __restrict__ b_lin,
    float* __restrict__ out)
{
  __shared__ float lds[16 * ROWW];
  const int lane = threadIdx.x;
  const int c16  = lane & 15;   // tile column / batch row in tile
  const int hh   = lane >> 4;   // lane half (selects K pair / row half)
  const int b0   = blockIdx.x * 16;

  for (int i = lane; i < 16 * ROWW; i += 32) lds[i] = 0.0f;
  if (lane < 16) lds[lane * ROWW + 11] = 1.0f;   // the "1.0" K slot
  __syncthreads();

  // ---- build persistent B operands (one-time, per-lane gathers) ----
  v2f B10[3], B11[3], B12[3], B13;
#pragma unroll
  for (int c = 0; c < 3; ++c) {
    int k0 = 4 * c + 2 * hh;
    B10[c].x = brow1(k0,     c16, 0, w_ih0, w_hh0, b_ih0, b_hh0);
    B10[c].y = brow1(k0 + 1, c16, 0, w_ih0, w_hh0, b_ih0, b_hh0);
    B11[c].x = brow1(k0,     c16, 1, w_ih0, w_hh0, b_ih0, b_hh0);
    B11[c].y = brow1(k0 + 1, c16, 1, w_ih0, w_hh0, b_ih0, b_hh0);
    B12[c].x = brow1(k0,     c16, 2, w_ih0, w_hh0, b_ih0, b_hh0);
    B12[c].y = brow1(k0 + 1, c16, 2, w_ih0, w_hh0, b_ih0, b_hh0);
  }
  B13.x = brow1(8 + 2 * hh, c16, 3, w_ih0, w_hh0, b_ih0, b_hh0);
  B13.y = brow1(9 + 2 * hh, c16, 3, w_ih0, w_hh0, b_ih0, b_hh0);

  v2f B20[6], B21[6], B22[4], B23[3];
#pragma unroll
  for (int c = 0; c < 6; ++c) {
    int k0 = 4 * c + 2 * hh;
    B20[c].x = brow2(k0,     c16, 0, w_ih1, w_hh1, b_ih1, b_hh1);
    B20[c].y = brow2(k0 + 1, c16, 0, w_ih1, w_hh1, b_ih1, b_hh1);
    B21[c].x = brow2(k0,     c16, 1, w_ih1, w_hh1, b_ih1, b_hh1);
    B21[c].y = brow2(k0 + 1, c16, 1, w_ih1, w_hh1, b_ih1, b_hh1);
  }
#pragma unroll
  for (int c = 0; c < 4; ++c) {
    int k0 = 4 * (c + 2) + 2 * hh;
    B22[c].x = brow2(k0,     c16, 2, w_ih1, w_hh1, b_ih1, b_hh1);
    B22[c].y = brow2(k0 + 1, c16, 2, w_ih1, w_hh1, b_ih1, b_hh1);
  }
#pragma unroll
  for (int c = 0; c < 3; ++c) {
    int k0 = 4 * c + 2 * hh;
    B23[c].x = brow2(k0,     c16, 3, w_ih1, w_hh1, b_ih1, b_hh1);
    B23[c].y = brow2(k0 + 1, c16, 3, w_ih1, w_hh1, b_ih1, b_hh1);
  }

  // head weights: uniform -> SGPRs
  float wl[HN];
#pragma unroll
  for (int j = 0; j < HN; ++j) wl[j] = w_lin[j];
  const float blin = b_lin[0];

  v8f h1, h2, zero8;
#pragma unroll
  for (int i = 0; i < 8; ++i) { h1[i] = 0.0f; h2[i] = 0.0f; zero8[i] = 0.0f; }

  const int rb = c16 * ROWW;
  const float* xrow = x   + (size_t)(b0 + c16) * TT;
  float*       orow = out + (size_t)(b0 + c16) * TT;

  // software-pipelined x loads (depth 2): xv0 = x[t], xv1 = x[t+1]
  float xv0 = xrow[0];
  float xv1 = xrow[1];

  for (int t = 0; t < TT; ++t) {
    // issue the load for t+2 now; its wait lands two iterations later
    int tp = t + 2; tp = (tp < TT) ? tp : (TT - 1);
    float xv2 = xrow[tp];

    // stage x_t (all 32 lanes: lanes 16..31 duplicate same addr/value)
    lds[rb + 10] = xv0;

    // ---- load all A chunks (chunks 3..5 = previous-step h2; chunk 2 = h1[8..9],x,1) ----
    v2f a0, a1, a2, a3, a4, a5;
    a0.x = lds[rb +  0 + 2 * hh];  a0.y = lds[rb +  1 + 2 * hh];
    a1.x = lds[rb +  4 + 2 * hh];  a1.y = lds[rb +  5 + 2 * hh];
    a2.x = lds[rb +  8 + 2 * hh];  a2.y = lds[rb +  9 + 2 * hh];
    a3.x = lds[rb + 12 + 2 * hh];  a3.y = lds[rb + 13 + 2 * hh];
    a4.x = lds[rb + 16 + 2 * hh];  a4.y = lds[rb + 17 + 2 * hh];
    a5.x = lds[rb + 20 + 2 * hh];  a5.y = lds[rb + 21 + 2 * hh];

    // ---------------- layer 1 WMMAs ----------------
    v8f s0 = wmma4(a0, B10[0], zero8); s0 = wmma4(a1, B10[1], s0); s0 = wmma4(a2, B10[2], s0);
    v8f s1 = wmma4(a0, B11[0], zero8); s1 = wmma4(a1, B11[1], s1); s1 = wmma4(a2, B11[2], s1);
    v8f s2 = wmma4(a0, B12[0], zero8); s2 = wmma4(a1, B12[1], s2); s2 = wmma4(a2, B12[2], s2);
    v8f s3 = wmma4(a2, B13, zero8);    // xn lives only in K slots 10/11

    // ---- layer 2, h2-side partials: independent of layer-1 result ----
    // (chunk 2's h1 rows have zero weights in tile 2, so the stale chunk is safe)
    v8f p0 = wmma4(a3, B20[3], zero8); p0 = wmma4(a4, B20[4], p0); p0 = wmma4(a5, B20[5], p0);
    v8f p1 = wmma4(a3, B21[3], zero8); p1 = wmma4(a4, B21[4], p1); p1 = wmma4(a5, B21[5], p1);
    v8f p2 = wmma4(a2, B22[0], zero8); p2 = wmma4(a3, B22[1], p2);
    p2 = wmma4(a4, B22[2], p2);        p2 = wmma4(a5, B22[3], p2);

    // ---------------- layer 1 gates ----------------
#pragma unroll
    for (int i = 0; i < 8; ++i) {
      float r  = __builtin_amdgcn_rcpf(1.0f + __builtin_amdgcn_exp2f(-s0[i]));
      float z  = __builtin_amdgcn_rcpf(1.0f + __builtin_amdgcn_exp2f(-s1[i]));
      float np = s3[i] + r * s2[i];
      float n  = 2.0f * __builtin_amdgcn_rcpf(1.0f + __builtin_amdgcn_exp2f(-np)) - 1.0f;
      h1[i] = n + z * (h1[i] - n);
    }
    if (c16 < HN) {                     // D-layout -> LDS row (transpose store)
#pragma unroll
      for (int i = 0; i < 8; ++i) lds[(i + 8 * hh) * ROWW + c16] = h1[i];
    }

    // ---------------- layer 2, h1-side (fresh chunks 0..2) ----------------
    v2f f0, f1, f2;
    f0.x = lds[rb + 0 + 2 * hh];  f0.y = lds[rb + 1 + 2 * hh];
    f1.x = lds[rb + 4 + 2 * hh];  f1.y = lds[rb + 5 + 2 * hh];
    f2.x = lds[rb + 8 + 2 * hh];  f2.y = lds[rb + 9 + 2 * hh];

    v8f t0 = wmma4(f0, B20[0], p0); t0 = wmma4(f1, B20[1], t0); t0 = wmma4(f2, B20[2], t0);
    v8f t1 = wmma4(f0, B21[0], p1); t1 = wmma4(f1, B21[1], t1); t1 = wmma4(f2, B21[2], t1);
    v8f t3 = wmma4(f0, B23[0], zero8); t3 = wmma4(f1, B23[1], t3); t3 = wmma4(f2, B23[2], t3);

    // ---------------- layer 2 gates ----------------
#pragma unroll
    for (int i = 0; i < 8; ++i) {
      float r  = __builtin_amdgcn_rcpf(1.0f + __builtin_amdgcn_exp2f(-t0[i]));
      float z  = __builtin_amdgcn_rcpf(1.0f + __builtin_amdgcn_exp2f(-t1[i]));
      float np = t3[i] + r * p2[i];
      float n  = 2.0f * __builtin_amdgcn_rcpf(1.0f + __builtin_amdgcn_exp2f(-np)) - 1.0f;
      h2[i] = n + z * (h2[i] - n);
    }
    if (c16 < HN) {
#pragma unroll
      for (int i = 0; i < 8; ++i) lds[(i + 8 * hh) * ROWW + 12 + c16] = h2[i];
    }

    // ---------------- linear head ----------------
    float acc = blin;
#pragma unroll
    for (int j = 0; j < HN; ++j) acc += lds[rb + 12 + j] * wl[j];
    orow[t] = acc;                      // lanes 16..31 duplicate same addr/value

    xv0 = xv1; xv1 = xv2;
  }
}

extern "C" void kernel_launch(void* const* d_in, const int* in_sizes, int n_in,
                              void* d_out, int out_size, void* d_ws, size_t ws_size,
                              hipStream_t stream) {
  const float* x     = (const float*)d_in[0];
  const float* w_ih0 = (const float*)d_in[1];
  const float* w_hh0 = (const float*)d_in[2];
  const float* b_ih0 = (const float*)d_in[3];
  const float* b_hh0 = (const float*)d_in[4];
  const float* w_ih1 = (const float*)d_in[5];
  const float* w_hh1 = (const float*)d_in[6];
  const float* b_ih1 = (const float*)d_in[7];
  const float* b_hh1 = (const float*)d_in[8];
  const float* w_lin = (const float*)d_in[9];
  const float* b_lin = (const float*)d_in[10];
  float* out = (float*)d_out;
  (void)in_sizes; (void)n_in; (void)out_size; (void)d_ws; (void)ws_size;

  gru2_wmma<<<dim3(BB / 16), dim3(32), 0, stream>>>(
      x, w_ih0, w_hh0, b_ih0, b_hh0, w_ih1, w_hh1, b_ih1, b_hh1, w_lin, b_lin, out);
}